// GraphAttentionEncoder_50749333569598
// MI455X (gfx1250) — compile-verified
//
#include <hip/hip_runtime.h>
#include <math.h>

#define HD 256
#define NN 20000
#define EE 320000
#define LL 2

#define MTILES (NN / 16)          // 1250
#define WS_STRIDE 260             // padded LDS row stride (floats): banks 4r, 16B aligned

typedef __attribute__((ext_vector_type(2))) float v2f;
typedef __attribute__((ext_vector_type(8))) float v8f;
typedef __attribute__((__vector_size__(16))) int v4i_t;

// ---------------- utility ----------------
__global__ void k_zero(float* __restrict__ p, int n) {
  int i = blockIdx.x * 256 + threadIdx.x;
  if (i < n) p[i] = 0.f;
}

__device__ __forceinline__ void block_reduce(float* red, int t, int op_sum) {
  __syncthreads();
  for (int off = 128; off > 0; off >>= 1) {
    if (t < off) red[t] = op_sum ? (red[t] + red[t + off]) : fmaxf(red[t], red[t + off]);
    __syncthreads();
  }
}

// ---------------- embedding gather ----------------
__global__ void k_embed(const int* __restrict__ ids, const float* __restrict__ emb,
                        float* __restrict__ ns) {
  int n = blockIdx.x, t = threadIdx.x;
  ns[(size_t)n * HD + t] = emb[(size_t)ids[n] * HD + t];
}

// ---------------- fp32 WMMA GEMM: D = A @ W^T (+optional tanh(x+bias)) ----------------
// Block = 8 waves. blockIdx.y = column tile (16 of them). Block stages the 16x256 W
// strip for its column tile into LDS (async-to-LDS when available), then each wave
// computes one 16x16 output tile (8 consecutive M-tiles per block).
__global__ void k_wmma_gemm(const float* __restrict__ A, const float* __restrict__ W,
                            const float* __restrict__ bias, float* __restrict__ D,
                            int tanh_ep) {
  __shared__ float WS[16 * WS_STRIDE];

  const int t = threadIdx.x;
  const int col0 = blockIdx.y * 16;

  // ---- stage W[col0:col0+16, 0:256] into LDS (16KB), b128 per thread x4 ----
  {
#if __has_builtin(__builtin_amdgcn_global_load_async_to_lds_b128)
    typedef __attribute__((address_space(1))) v4i_t* gv4_t;
    typedef __attribute__((address_space(3))) v4i_t* lv4_t;
#pragma unroll
    for (int i = 0; i < 4; ++i) {
      int q = t + i * 256;            // float4 index 0..1023
      int row = q >> 6;               // 64 float4 per row
      int kq = (q & 63) << 2;         // float offset in row
      const v4i_t* g = (const v4i_t*)(W + (size_t)(col0 + row) * HD + kq);
      v4i_t* l = (v4i_t*)&WS[row * WS_STRIDE + kq];
      __builtin_amdgcn_global_load_async_to_lds_b128((gv4_t)g, (lv4_t)l, 0, 0);
    }
#if __has_builtin(__builtin_amdgcn_s_wait_asynccnt)
    __builtin_amdgcn_s_wait_asynccnt(0);
#else
    asm volatile("s_wait_asynccnt 0" ::: "memory");
#endif
#else
#pragma unroll
    for (int i = 0; i < 4; ++i) {
      int q = t + i * 256;
      int row = q >> 6;
      int kq = (q & 63) << 2;
      *(float4*)&WS[row * WS_STRIDE + kq] =
          *(const float4*)(W + (size_t)(col0 + row) * HD + kq);
    }
#endif
  }
  __syncthreads();

  const int wave = t >> 5;
  const int lane = t & 31;
  const int tileM = blockIdx.x * 8 + wave;
  if (tileM < MTILES) {               // wave-uniform guard: EXEC stays all-ones inside
    const int row0 = tileM * 16;
    const int lm = lane & 15;
    const int half = lane >> 4;

    // fp32 16x16x4 fragment layout: vgpr pair = K {0,1} (lanes 0-15) / {2,3} (16-31)
    const float* arow = A + (size_t)(row0 + lm) * HD + 2 * half;
    const float* brow = &WS[lm * WS_STRIDE + 2 * half];   // B[k][n] = W[n][k] from LDS

    v8f c = {0.f, 0.f, 0.f, 0.f, 0.f, 0.f, 0.f, 0.f};
#pragma unroll 8
    for (int kk = 0; kk < HD; kk += 4) {
      v2f a, b;
      a.x = arow[kk]; a.y = arow[kk + 1];
      b.x = brow[kk]; b.y = brow[kk + 1];
      c = __builtin_amdgcn_wmma_f32_16x16x4_f32(false, a, false, b, (short)0, c, false, false);
    }
#pragma unroll
    for (int r = 0; r < 8; ++r) {
      int m = row0 + r + 8 * half;    // D layout: vgpr r -> M = r (+8 for upper half)
      int n = col0 + lm;
      float v = c[r];
      if (tanh_ep) v = tanhf(v + bias[n]);
      D[(size_t)m * HD + n] = v;
    }
  }
}

// ---------------- per-node edge-MLP dot factorization ----------------
// a_src[n] = h[n] . ew[0:H], a_dst[n] = h[n] . ew[H:2H]
__global__ void k_node_dots(const float* __restrict__ h, const float* __restrict__ ew,
                            float* __restrict__ a_src, float* __restrict__ a_dst) {
  __shared__ float s1[256];
  __shared__ float s2[256];
  int n = blockIdx.x, t = threadIdx.x;
  float x = h[(size_t)n * HD + t];
  s1[t] = x * ew[t];
  s2[t] = x * ew[HD + t];
  __syncthreads();
  for (int off = 128; off > 0; off >>= 1) {
    if (t < off) { s1[t] += s1[t + off]; s2[t] += s2[t + off]; }
    __syncthreads();
  }
  if (t == 0) { a_src[n] = s1[0]; a_dst[n] = s2[0]; }
}

// ---------------- edge scores (leaky relu) + fused block max ----------------
__global__ void k_edge_scores(const int* __restrict__ src, const int* __restrict__ dst,
                              const float* __restrict__ a_src, const float* __restrict__ a_dst,
                              float* __restrict__ scores, float* __restrict__ pmax, int E) {
  __shared__ float red[256];
  int e = blockIdx.x * 256 + threadIdx.x;
  float v = -INFINITY;
  if (e < E) {
    float s = a_src[src[e]] + a_dst[dst[e]];
    s = (s >= 0.f) ? s : 0.2f * s;   // leaky_relu(0.2)
    scores[e] = s;
    v = s;
  }
  red[threadIdx.x] = v;
  block_reduce(red, threadIdx.x, 0);
  if (threadIdx.x == 0) pmax[blockIdx.x] = red[0];
}

__global__ void k_partial_max(const float* __restrict__ in, int n, float* __restrict__ out) {
  __shared__ float red[256];
  int i = blockIdx.x * 256 + threadIdx.x;
  red[threadIdx.x] = (i < n) ? in[i] : -INFINITY;
  block_reduce(red, threadIdx.x, 0);
  if (threadIdx.x == 0) out[blockIdx.x] = red[0];
}

__global__ void k_final_reduce(const float* __restrict__ in, int n, float* __restrict__ out,
                               int op_sum) {
  __shared__ float red[256];
  int t = threadIdx.x;
  float v = op_sum ? 0.f : -INFINITY;
  for (int i = t; i < n; i += 256) v = op_sum ? (v + in[i]) : fmaxf(v, in[i]);
  red[t] = v;
  block_reduce(red, t, op_sum);
  if (t == 0) out[0] = red[0];
}

// ---------------- exp(score - max) and denominator segment-sum ----------------
__global__ void k_edge_exp(float* __restrict__ scores, const int* __restrict__ dst,
                           const float* __restrict__ mx, float* __restrict__ denom, int E) {
  int e = blockIdx.x * 256 + threadIdx.x;
  if (e < E) {
    float a = expf(scores[e] - mx[0]);
    scores[e] = a;                         // scores buffer becomes attn
    atomicAdd(&denom[dst[e]], a);
  }
}

// ---------------- message aggregation: agg[dst] += h[src] * w ----------------
__global__ void k_edge_agg(const int* __restrict__ src, const int* __restrict__ dst,
                           const float* __restrict__ h, const float* __restrict__ attn,
                           const float* __restrict__ denom, float* __restrict__ agg) {
  int e = blockIdx.x;
  int t = threadIdx.x;
  int s = src[e], d = dst[e];
  float w = attn[e] / (denom[d] + 1e-6f);
  atomicAdd(&agg[(size_t)d * HD + t], h[(size_t)s * HD + t] * w);
}

// ---------------- residual + layernorm + exact gelu, in-place on ns ----------------
__global__ void k_ln_gelu(const float* __restrict__ agg, float* __restrict__ ns,
                          const float* __restrict__ g, const float* __restrict__ b) {
  __shared__ float red[256];
  __shared__ float mean_s, rstd_s;
  int n = blockIdx.x, t = threadIdx.x;
  float x = agg[(size_t)n * HD + t] + ns[(size_t)n * HD + t];
  red[t] = x;
  block_reduce(red, t, 1);
  if (t == 0) mean_s = red[0] * (1.f / HD);
  __syncthreads();
  float dx = x - mean_s;
  red[t] = dx * dx;
  block_reduce(red, t, 1);
  if (t == 0) rstd_s = rsqrtf(red[0] * (1.f / HD) + 1e-5f);
  __syncthreads();
  float y = dx * rstd_s * g[t] + b[t];
  y = 0.5f * y * (1.f + erff(y * 0.70710678118654752f));   // exact gelu
  ns[(size_t)n * HD + t] = y;
}

// ---------------- pooling: s[n] = proj[n] . score_w, masked ----------------
__global__ void k_pool_score(const float* __restrict__ proj, const float* __restrict__ sw,
                             const int* __restrict__ ids, float* __restrict__ s) {
  __shared__ float red[256];
  int n = blockIdx.x, t = threadIdx.x;
  red[t] = proj[(size_t)n * HD + t] * sw[t];
  block_reduce(red, t, 1);
  if (t == 0) s[n] = (ids[n] != 0) ? red[0] : -INFINITY;
}

__global__ void k_exp_psum(float* __restrict__ s, const float* __restrict__ mx,
                           float* __restrict__ psum, int n) {
  __shared__ float red[256];
  int i = blockIdx.x * 256 + threadIdx.x;
  float e = 0.f;
  if (i < n) {
    float m = mx[0];
    float v = (m > -3.0e38f) ? expf(s[i] - m) : 0.f;  // all-masked -> zeros (nan_to_num)
    s[i] = v;
    e = v;
  }
  red[threadIdx.x] = e;
  block_reduce(red, threadIdx.x, 1);
  if (threadIdx.x == 0) psum[blockIdx.x] = red[0];
}

__global__ void k_finalize_w(const float* __restrict__ s, const float* __restrict__ ssum,
                             float* __restrict__ wout, int n) {
  int i = blockIdx.x * 256 + threadIdx.x;
  if (i < n) {
    float d = ssum[0];
    wout[i] = (d > 0.f) ? s[i] / d : 0.f;
  }
}

// ---------------- graph_vec = w @ ns ----------------
__global__ void k_graphvec(const float* __restrict__ w, const float* __restrict__ ns,
                           float* __restrict__ gvec) {
  int t = threadIdx.x;
  int base = blockIdx.x * 32;            // 625 blocks * 32 nodes = 20000
  float acc = 0.f;
#pragma unroll 4
  for (int i = 0; i < 32; ++i) {
    int n = base + i;
    acc += w[n] * ns[(size_t)n * HD + t];
  }
  atomicAdd(&gvec[t], acc);
}

// ===================== host launcher =====================
extern "C" void kernel_launch(void* const* d_in, const int* in_sizes, int n_in,
                              void* d_out, int out_size, void* d_ws, size_t ws_size,
                              hipStream_t stream) {
  const int*   node_ids = (const int*)d_in[0];
  const int*   edge_idx = (const int*)d_in[1];   // [2,E]: row0 src, row1 dst
  const float* emb      = (const float*)d_in[2];
  const float* proj_w   = (const float*)d_in[3]; // [L,H,H]
  const float* edge_w   = (const float*)d_in[4]; // [L,2H]
  const float* ln_g     = (const float*)d_in[5]; // [L,H]
  const float* ln_b     = (const float*)d_in[6]; // [L,H]
  const float* pool_w   = (const float*)d_in[7]; // [H,H]
  const float* pool_b   = (const float*)d_in[8]; // [H]
  const float* score_w  = (const float*)d_in[9]; // [H]

  const int* src = edge_idx;
  const int* dst = edge_idx + EE;

  const size_t NH = (size_t)NN * HD;
  float* ws    = (float*)d_ws;
  float* ns    = ws;                 // [N,H]
  float* hbuf  = ws + NH;            // [N,H] (proj h, later pooling proj)
  float* agg   = ws + 2 * NH;        // [N,H]
  float* a_src = ws + 3 * NH;        // [N]
  float* a_dst = a_src + NN;         // [N]
  float* denom = a_dst + NN;         // [N]
  float* svec  = denom + NN;         // [N]
  float* score = svec + NN;          // [E] (becomes attn)
  float* part  = score + EE;         // [2048] partial reductions
  float* smax  = part + 2048;        // [1]
  float* ssum  = smax + 1;           // [1]

  float* out   = (float*)d_out;      // [H + N]: graph_vec then w
  float* w_out = out + HD;

  const dim3 GEMM_GRID((MTILES + 7) / 8, HD / 16);   // 157 x 16, 8 M-tiles/block
  const int EB = EE / 256;                           // 1250 (exact)
  const int NB = (NN + 255) / 256;                   // 79

  // ns = emb[node_ids]
  k_embed<<<NN, 256, 0, stream>>>(node_ids, emb, ns);

  for (int l = 0; l < LL; ++l) {
    // zero agg + denom
    k_zero<<<(int)((NH + 255) / 256), 256, 0, stream>>>(agg, (int)NH);
    k_zero<<<NB, 256, 0, stream>>>(denom, NN);

    // h = ns @ W_l^T   (fp32 WMMA, W strip staged in LDS)
    k_wmma_gemm<<<GEMM_GRID, 256, 0, stream>>>(ns, proj_w + (size_t)l * HD * HD,
                                               nullptr, hbuf, 0);
    // per-node halves of the edge MLP
    k_node_dots<<<NN, 256, 0, stream>>>(hbuf, edge_w + (size_t)l * 2 * HD, a_src, a_dst);
    // edge scores + global max
    k_edge_scores<<<EB, 256, 0, stream>>>(src, dst, a_src, a_dst, score, part, EE);
    k_final_reduce<<<1, 256, 0, stream>>>(part, EB, smax, 0);
    // attn = exp(score - max); denom = segment_sum(attn, dst)
    k_edge_exp<<<EB, 256, 0, stream>>>(score, dst, smax, denom, EE);
    // agg[dst] += h[src] * attn/(denom+eps)
    k_edge_agg<<<EE, 256, 0, stream>>>(src, dst, hbuf, score, denom, agg);
    // ns = gelu(layernorm(agg + ns))
    k_ln_gelu<<<NN, 256, 0, stream>>>(agg, ns, ln_g + (size_t)l * HD, ln_b + (size_t)l * HD);
  }

  // pooling: proj = tanh(ns @ pool_w^T + pool_b)   (fp32 WMMA, fused epilogue)
  k_wmma_gemm<<<GEMM_GRID, 256, 0, stream>>>(ns, pool_w, pool_b, hbuf, 1);
  // s[n] = proj[n].score_w, masked with -inf
  k_pool_score<<<NN, 256, 0, stream>>>(hbuf, score_w, node_ids, svec);
  // softmax over nodes
  k_partial_max<<<NB, 256, 0, stream>>>(svec, NN, part);
  k_final_reduce<<<1, 256, 0, stream>>>(part, NB, smax, 0);
  k_exp_psum<<<NB, 256, 0, stream>>>(svec, smax, part, NN);
  k_final_reduce<<<1, 256, 0, stream>>>(part, NB, ssum, 1);
  k_finalize_w<<<NB, 256, 0, stream>>>(svec, ssum, w_out, NN);
  // graph_vec = w @ ns
  k_zero<<<1, 256, 0, stream>>>(out, HD);
  k_graphvec<<<NN / 32, 256, 0, stream>>>(w_out, ns, out);
}